// LovaszSoftmax_5222680232275
// MI455X (gfx1250) — compile-verified
//
#include <hip/hip_runtime.h>
#include <hip/hip_bf16.h>
#include <math.h>

#define NCLS  20
#define BINS  65536
#define HW    (512*1024)       /* 2^19 */
#define BATCH 4
#define NPIX  (BATCH*HW)

typedef float v2f __attribute__((ext_vector_type(2)));
typedef float v8f __attribute__((ext_vector_type(8)));
typedef unsigned long long u64;

// ---------------- workspace layout (u32 words) ----------------
// cfg  u64[NCLS][BINS]   packed {fg count (hi32), bin count (lo32)}
// efx  u64[NCLS][BINS]   error sum, 2^32 fixed point (deterministic)
// gts  u32[NCLS]         total fg per class
// acc  f32[2]            {loss numerator, present-class count}
#define CFG_OFF  0
#define EFX_OFF  (2*NCLS*BINS)
#define GTS_OFF  (4*NCLS*BINS)
#define ACC_OFF  (4*NCLS*BINS + NCLS)
#define WS_WORDS (ACC_OFF + 2)          /* ~21 MB */

__global__ void k_zero(unsigned* __restrict__ ws, int words) {
    for (int i = blockIdx.x * blockDim.x + threadIdx.x; i < words;
         i += gridDim.x * blockDim.x)
        ws[i] = 0u;
}

// Pass 1: fused softmax + per-class error histogram.
// Reads 167.8MB logits once (7.6us HBM floor); the real cost is the L2
// atomics, so (count,fg) are packed into ONE u64 atomic and the error sum
// uses a u64 fixed-point atomic: 2 atomics per (pixel,class) instead of 3.
__global__ void k_hist(const float* __restrict__ logits,
                       const int*   __restrict__ labels,
                       u64*      __restrict__ cfg,
                       u64*      __restrict__ efx,
                       unsigned* __restrict__ gts) {
    int t = blockIdx.x * blockDim.x + threadIdx.x;
    if (t >= NPIX) return;
    int b  = t >> 19;            // / HW
    int hw = t & (HW - 1);
    const float* lp = logits + (size_t)b * NCLS * HW + hw;

    float v[NCLS];
    float m = -INFINITY;
#pragma unroll
    for (int c = 0; c < NCLS; ++c) { v[c] = lp[(size_t)c * HW]; m = fmaxf(m, v[c]); }
    float s = 0.f;
#pragma unroll
    for (int c = 0; c < NCLS; ++c) { v[c] = __expf(v[c] - m); s += v[c]; }
    float inv = 1.0f / s;

    int  lab   = labels[t];
    bool valid = (lab != 0);                 // IGNORE == 0
    if (valid) atomicAdd(&gts[lab], 1u);

    // class 0 can never be 'present' (fg requires lab==0 && lab!=0) -> skip it
#pragma unroll
    for (int c = 1; c < NCLS; ++c) {
        float pred = valid ? v[c] * inv : 0.0f;
        bool  isfg = (valid && lab == c);
        float err  = fabsf((isfg ? 1.0f : 0.0f) - pred);
        unsigned key = (unsigned)fminf(err * 65536.0f, 65535.0f);
        unsigned p   = (BINS - 1u) - key;    // position in DESCENDING error order
        unsigned idx = (unsigned)(c * BINS) + p;
        u64 packed = 1ull | ((u64)(isfg ? 1u : 0u) << 32);
        atomicAdd(&cfg[idx], packed);
        atomicAdd(&efx[idx], (u64)(err * 4294967296.0f));   // err * 2^32
    }
}

// Pass 2: per-class ordered prefix scan of (count, fg) + Lovasz telescoped
// dot product. One wave per class. The 256-bin chunk scan is done with
// V_WMMA_F32_16X16X4_F32: D = X(16x16) * UpperTriOnes(16x16) via 4 chained
// K=4 WMMAs -> exact f32 row-wise inclusive prefix sums (counts < 2^24).
__global__ void __launch_bounds__(32)
k_scan(const u64*      __restrict__ cfg,
       const u64*      __restrict__ efx,
       const unsigned* __restrict__ gts,
       float* __restrict__ acc) {
    __shared__ float sN[256];   // counts chunk as 16x16 row-major
    __shared__ float sK[256];   // fg chunk
    __shared__ float rTN[16], rTF[16];   // row totals
    __shared__ float pTN[16], pTF[16];   // exclusive prefix of row totals

    int c = blockIdx.x;
    float G = (float)gts[c];
    if (G <= 0.0f) return;      // class not present (uniform branch, EXEC stays full)

    const int L  = threadIdx.x;
    const int M  = L & 15;      // A-matrix row
    const int hi = L >> 4;
    const int Nn = L & 15;      // B/D column

    float carryN = 0.f, carryF = 0.f, lloss = 0.f;

    for (int chunk = 0; chunk < BINS / 256; ++chunk) {
        int base = c * BINS + chunk * 256;
        // stage chunk into LDS as floats (exact: counts < 2^24)
#pragma unroll
        for (int q = 0; q < 8; ++q) {
            int idx = L * 8 + q;
            u64 pv  = cfg[base + idx];
            sN[idx] = (float)(unsigned)pv;          // bin count
            sK[idx] = (float)(unsigned)(pv >> 32);  // fg count
        }
        __syncthreads();

        // row-wise inclusive prefix of the 16x16 tiles via WMMA:
        // D[i][j] = sum_{k<=j} X[i][k]  (T upper-triangular ones)
        v8f Dn = {}; v8f Df = {};
#pragma unroll
        for (int k0 = 0; k0 < 16; k0 += 4) {
            v2f a, af, bt;
            a.x  = sN[16 * M + k0 + 2 * hi + 0];   // A 16x4: lanes0-15 K0/K1, 16-31 K2/K3
            a.y  = sN[16 * M + k0 + 2 * hi + 1];
            af.x = sK[16 * M + k0 + 2 * hi + 0];
            af.y = sK[16 * M + k0 + 2 * hi + 1];
            bt.x = (k0 + hi     <= Nn) ? 1.0f : 0.0f;  // T[k][j] = (k<=j)
            bt.y = (k0 + 2 + hi <= Nn) ? 1.0f : 0.0f;
            Dn = __builtin_amdgcn_wmma_f32_16x16x4_f32(false, a,  false, bt,
                                                       (short)0, Dn, false, false);
            Df = __builtin_amdgcn_wmma_f32_16x16x4_f32(false, af, false, bt,
                                                       (short)0, Df, false, false);
        }

        // row totals live at column 15 (lanes 15 and 31)
        if (Nn == 15) {
#pragma unroll
            for (int r = 0; r < 8; ++r) {
                rTN[r + 8 * hi] = Dn[r];
                rTF[r + 8 * hi] = Df[r];
            }
        }
        __syncthreads();
        if (L < 16) {
            float on = 0.f, of = 0.f;
            for (int rr = 0; rr < L; ++rr) { on += rTN[rr]; of += rTF[rr]; }
            pTN[L] = on; pTF[L] = of;
        }
        __syncthreads();

        // telescoped Lovasz contribution per non-empty bin:
        // e_mean * (J(P+n, F+k) - J(P, F)),  J(P,F) = 1 - (G-F)/(G+P-F)
#pragma unroll
        for (int r = 0; r < 8; ++r) {
            int i  = r + 8 * hi;            // tile row
            int lp = 16 * i + Nn;           // linear pos in chunk
            float n = sN[lp], k = sK[lp];
            if (n > 0.f) {
                float P0 = carryN + pTN[i] + Dn[r] - n;   // exclusive prefixes
                float F0 = carryF + pTF[i] + Df[r] - k;
                float P1 = P0 + n, F1 = F0 + k;
                float J0 = 1.0f - (G - F0) / (G + P0 - F0);  // J(0)=0 falls out
                float J1 = 1.0f - (G - F1) / (G + P1 - F1);
                float se = (float)efx[base + lp] * 2.3283064365386963e-10f; // *2^-32
                lloss += (se / n) * (J1 - J0);
            }
        }
        carryN += pTN[15] + rTN[15];
        carryF += pTF[15] + rTF[15];
        __syncthreads();
    }

    // wave32 reduction
#pragma unroll
    for (int o = 16; o > 0; o >>= 1) lloss += __shfl_down(lloss, o);
    if (L == 0) {
        atomicAdd(&acc[0], lloss);
        atomicAdd(&acc[1], 1.0f);
    }
}

__global__ void k_final(const float* __restrict__ acc, float* __restrict__ out) {
    out[0] = acc[0] / fmaxf(acc[1], 1.0f);
}

extern "C" void kernel_launch(void* const* d_in, const int* in_sizes, int n_in,
                              void* d_out, int out_size, void* d_ws, size_t ws_size,
                              hipStream_t stream) {
    const float* logits = (const float*)d_in[0];
    const int*   labels = (const int*)d_in[1];
    unsigned* ws  = (unsigned*)d_ws;           // needs ~21 MB
    u64*      cfg = (u64*)(ws + CFG_OFF);
    u64*      efx = (u64*)(ws + EFX_OFF);
    unsigned* gts = ws + GTS_OFF;
    float*    acc = (float*)(ws + ACC_OFF);

    hipLaunchKernelGGL(k_zero, dim3(2048), dim3(256), 0, stream, ws, (int)WS_WORDS);
    hipLaunchKernelGGL(k_hist, dim3((NPIX + 255) / 256), dim3(256), 0, stream,
                       logits, labels, cfg, efx, gts);
    hipLaunchKernelGGL(k_scan, dim3(NCLS), dim3(32), 0, stream,
                       cfg, efx, gts, acc);
    hipLaunchKernelGGL(k_final, dim3(1), dim3(1), 0, stream, acc, (float*)d_out);
}